// YOLO_loss_83545703842405
// MI455X (gfx1250) — compile-verified
//
#include <hip/hip_runtime.h>
#include <stdint.h>

#define SGRID   7
#define NFEAT   30                 // 5*B + C
#define NCELLS  (8192 * SGRID * SGRID)   // 401408
#define TPB     256
#define NBLK    (NCELLS / TPB)     // 1568, exact
#define CELL_B  (NFEAT * 4)        // 120 bytes per cell
#define WAVE_F  (32 * NFEAT)       // 960 floats per wave tile

__global__ __launch_bounds__(TPB)
void yolo_loss_partial(const float* __restrict__ pred,
                       const float* __restrict__ targ,
                       float* __restrict__ part)
{
    // per-wave staging tiles: 8 waves * 2 arrays * 3840B = 61440B
    __shared__ __align__(16) float tile[TPB / 32][2][WAVE_F];
    __shared__ float red[TPB];

    const int tid  = threadIdx.x;
    const int lane = tid & 31;
    const int w    = tid >> 5;

    const unsigned waveCell = (unsigned)blockIdx.x * TPB + ((unsigned)w << 5);
    const unsigned gbyte    = waveCell * (unsigned)CELL_B;   // < 2^26, fits i32

    const unsigned ldsPred = (unsigned)(uint64_t)(uintptr_t)&tile[w][0][0];
    const unsigned ldsTarg = (unsigned)(uint64_t)(uintptr_t)&tile[w][1][0];
    const uint64_t predBase = (uint64_t)(uintptr_t)pred;
    const uint64_t targBase = (uint64_t)(uintptr_t)targ;

    // 3840 bytes per array per wave: 7 full rounds of 32 lanes x 16B,
    // plus one half round (lanes 0..15). All transfers 16B-aligned, coalesced.
    #pragma unroll
    for (int r = 0; r < 8; ++r) {
        if (r < 7 || lane < 16) {
            const unsigned off = (unsigned)(r * 512 + lane * 16);
            const unsigned gp  = gbyte + off;
            const unsigned lp  = ldsPred + off;
            const unsigned lt  = ldsTarg + off;
            asm volatile("global_load_async_to_lds_b128 %0, %1, %2 offset:0"
                         :: "v"(lp), "v"(gp), "s"(predBase) : "memory");
            asm volatile("global_load_async_to_lds_b128 %0, %1, %2 offset:0"
                         :: "v"(lt), "v"(gp), "s"(targBase) : "memory");
        }
    }
    asm volatile("s_wait_asynccnt 0" ::: "memory");

    const float* pc = &tile[w][0][lane * NFEAT];
    const float* tc = &tile[w][1][lane * NFEAT];

    float p[10], t[10];
    #pragma unroll
    for (int i = 0; i < 10; ++i) { p[i] = pc[i]; t[i] = tc[i]; }

    const float invS = 1.0f / (float)SGRID;
    float px1[2], py1[2], px2[2], py2[2], pA[2];
    float tx1[2], ty1[2], tx2[2], ty2[2], tA[2];
    #pragma unroll
    for (int b = 0; b < 2; ++b) {
        float x = p[5*b] * invS, y = p[5*b+1] * invS;
        float hw = 0.5f * p[5*b+2], hh = 0.5f * p[5*b+3];
        px1[b] = x - hw; px2[b] = x + hw; py1[b] = y - hh; py2[b] = y + hh;
        pA[b] = (px2[b] - px1[b]) * (py2[b] - py1[b]);
        x = t[5*b] * invS; y = t[5*b+1] * invS;
        hw = 0.5f * t[5*b+2]; hh = 0.5f * t[5*b+3];
        tx1[b] = x - hw; tx2[b] = x + hw; ty1[b] = y - hh; ty2[b] = y + hh;
        tA[b] = (tx2[b] - tx1[b]) * (ty2[b] - ty1[b]);
    }

    float iou[2][2];
    #pragma unroll
    for (int i = 0; i < 2; ++i) {
        #pragma unroll
        for (int j = 0; j < 2; ++j) {
            float lx = fmaxf(px1[i], tx1[j]), rx = fminf(px2[i], tx2[j]);
            float ly = fmaxf(py1[i], ty1[j]), ry = fminf(py2[i], ty2[j]);
            float iw = fmaxf(rx - lx, 0.0f), ih = fmaxf(ry - ly, 0.0f);
            float inter = iw * ih;
            iou[i][j] = inter / (pA[i] + tA[j] - inter);
        }
    }

    // per target box j: best pred box (argmax over i, first-index tie-break)
    int   idx[2]; float mi[2];
    #pragma unroll
    for (int j = 0; j < 2; ++j) {
        idx[j] = (iou[1][j] > iou[0][j]) ? 1 : 0;
        mi[j]  = fmaxf(iou[0][j], iou[1][j]);
    }
    bool  resp[2] = {false, false};
    float conf[2] = {0.0f, 0.0f};
    #pragma unroll
    for (int k = 0; k < 2; ++k) { resp[idx[k]] = true; conf[idx[k]] = mi[k]; }

    const bool obj = t[4] > 0.0f;

    float lloc = 0.0f, lobj = 0.0f, lnoobj = 0.0f;
    #pragma unroll
    for (int b = 0; b < 2; ++b) {
        float m  = (resp[b] && obj) ? 1.0f : 0.0f;
        float dx = p[5*b]   - t[5*b];
        float dy = p[5*b+1] - t[5*b+1];
        float dw = sqrtf(p[5*b+2]) - sqrtf(t[5*b+2]);
        float dh = sqrtf(p[5*b+3]) - sqrtf(t[5*b+3]);
        lloc += m * (dx*dx + dy*dy + dw*dw + dh*dh);
        float dc = p[5*b+4] - conf[b];
        lobj += m * dc * dc;
        lnoobj += p[5*b+4] * p[5*b+4];
    }

    float lcls = 0.0f;
    #pragma unroll
    for (int j = 10; j < 30; ++j) { float d = pc[j] - tc[j]; lcls += d * d; }

    float cell = obj ? (5.0f * lloc + lobj + lcls) : (0.5f * lnoobj);

    // block tree reduction -> one partial per block
    red[tid] = cell;
    __syncthreads();
    #pragma unroll
    for (int s = TPB / 2; s > 0; s >>= 1) {
        if (tid < s) red[tid] += red[tid + s];
        __syncthreads();
    }
    if (tid == 0) part[blockIdx.x] = red[0];
}

__global__ __launch_bounds__(256)
void yolo_loss_finish(const float* __restrict__ part, float* __restrict__ out)
{
    __shared__ float red[256];
    float s = 0.0f;
    for (int i = threadIdx.x; i < NBLK; i += 256) s += part[i];
    red[threadIdx.x] = s;
    __syncthreads();
    #pragma unroll
    for (int k = 128; k > 0; k >>= 1) {
        if (threadIdx.x < k) red[threadIdx.x] += red[threadIdx.x + k];
        __syncthreads();
    }
    if (threadIdx.x == 0) out[0] = red[0] * (1.0f / 8192.0f);
}

extern "C" void kernel_launch(void* const* d_in, const int* in_sizes, int n_in,
                              void* d_out, int out_size, void* d_ws, size_t ws_size,
                              hipStream_t stream) {
    (void)in_sizes; (void)n_in; (void)out_size; (void)ws_size;
    const float* pred = (const float*)d_in[0];
    const float* targ = (const float*)d_in[1];
    float* ws = (float*)d_ws;           // NBLK floats of scratch (6272 B)
    yolo_loss_partial<<<NBLK, TPB, 0, stream>>>(pred, targ, ws);
    yolo_loss_finish<<<1, 256, 0, stream>>>(ws, (float*)d_out);
}